// AleatoricCrossEntropyLoss_32865089749247
// MI455X (gfx1250) — compile-verified
//
#include <hip/hip_runtime.h>
#include <math.h>

// ---------------------------------------------------------------------------
// AleatoricCrossEntropyLoss for MI455X (gfx1250, wave32).
// Inputs : logit_var [512,1] f32, pred [512,3000] f32, true [512,3000] f32,
//          noise [25,512,3000] f32
// Outputs: 4 f32 scalars (gce_loss, variance_loss, undistorted_loss,
//          variance_depressor)
// Strategy: HBM-bound streaming row reductions; one workgroup per (t,b) row.
// Cross-lane sums use V_WMMA_F32_16X16X4_F32 with an all-ones B matrix.
// ---------------------------------------------------------------------------

typedef __attribute__((ext_vector_type(2))) float v2f;
typedef __attribute__((ext_vector_type(4))) float v4f;
typedef __attribute__((ext_vector_type(8))) float v8f;

#define B_DIM 512
#define C_DIM 3000
#define T_DIM 25
#define NV4   (C_DIM / 4)      // 750 float4 per row
#define BLOCK 256
#define NWAVE (BLOCK / 32)     // 8
#define MAXIT 3                // ceil(750/256)

// Full 32-lane sum via WMMA: A[m][0]=p_m, A[m][2]=p_{m+16}, others 0; B=ones.
// D[m][n] = p_m + p_{m+16}. Per lane, summing the 8 D VGPRs gives half-sums
// (rows 0-7 in lanes 0-15, rows 8-15 in lanes 16-31); xor-16 completes it.
__device__ __forceinline__ float wave_sum_wmma(float p) {
  v2f a; a.x = p;    a.y = 0.0f;
  v2f b; b.x = 1.0f; b.y = 1.0f;   // K=1,3 slots multiply A zeros -> harmless
  v8f c = {};
  c = __builtin_amdgcn_wmma_f32_16x16x4_f32(
      /*neg_a=*/false, a, /*neg_b=*/false, b,
      /*c_mod=*/(short)0, c, /*reuse_a=*/false, /*reuse_b=*/false);
  float s = c[0] + c[1] + c[2] + c[3] + c[4] + c[5] + c[6] + c[7];
  s += __shfl_xor(s, 16, 32);
  return s;
}

// 256-thread block sum; result valid in all threads. sbuf: 8 floats shared.
__device__ __forceinline__ float block_sum_256(float v, float* sbuf) {
  v = wave_sum_wmma(v);
  const int wave = threadIdx.x >> 5;
  const int lane = threadIdx.x & 31;
  __syncthreads();                  // protect sbuf reuse across calls
  if (lane == 0) sbuf[wave] = v;
  __syncthreads();
  float r = 0.0f;
#pragma unroll
  for (int w = 0; w < NWAVE; ++w) r += sbuf[w];
  return r;
}

// One workgroup per row. Rows [0, T*B): distorted logits pred + std_b*noise.
// Rows [T*B, T*B+B): undistorted (std = 0). Writes (lse*S - dot)/10 to ws[row].
__global__ __launch_bounds__(BLOCK) void row_kernel(
    const float* __restrict__ logit_var,
    const float* __restrict__ pred,
    const float* __restrict__ tru,
    const float* __restrict__ noise,
    float* __restrict__ ws) {
  const int row = blockIdx.x;
  const int tid = threadIdx.x;

  const bool distorted = row < T_DIM * B_DIM;
  int t, b;
  if (distorted) { t = row / B_DIM; b = row - t * B_DIM; }
  else           { t = 0;           b = row - T_DIM * B_DIM; }

  const float stdb = distorted ? sqrtf(logit_var[b]) : 0.0f;

  const v4f* p4 = (const v4f*)(pred  + (size_t)b * C_DIM);
  const v4f* t4 = (const v4f*)(tru   + (size_t)b * C_DIM);
  const v4f* n4 = (const v4f*)(noise + ((size_t)t * B_DIM + b) * C_DIM);

  v4f lx[MAXIT], lt[MAXIT];
  float lmax = -1.0e30f;

#pragma unroll
  for (int k = 0; k < MAXIT; ++k) {
    const int i = tid + k * BLOCK;
    v4f x  = {-1.0e30f, -1.0e30f, -1.0e30f, -1.0e30f};
    v4f tv = {0.0f, 0.0f, 0.0f, 0.0f};
    if (i < NV4) {
      const v4f pv = p4[i];
      tv = t4[i];
      // noise is single-use (153.6 MB stream): non-temporal so pred/true
      // (12 MB working set) stay resident in the 192 MB L2.
      const v4f nv = __builtin_nontemporal_load(n4 + i);
      x.x = fmaf(stdb, nv.x, pv.x);
      x.y = fmaf(stdb, nv.y, pv.y);
      x.z = fmaf(stdb, nv.z, pv.z);
      x.w = fmaf(stdb, nv.w, pv.w);
    }
    lx[k] = x; lt[k] = tv;
    lmax = fmaxf(lmax, fmaxf(fmaxf(x.x, x.y), fmaxf(x.z, x.w)));
  }

  // Block max (shuffles; WMMA cannot do max).
#pragma unroll
  for (int off = 16; off > 0; off >>= 1)
    lmax = fmaxf(lmax, __shfl_xor(lmax, off, 32));
  __shared__ float smax[NWAVE];
  const int wave = tid >> 5, lane = tid & 31;
  if (lane == 0) smax[wave] = lmax;
  __syncthreads();
#pragma unroll
  for (int w = 0; w < NWAVE; ++w) lmax = fmaxf(lmax, smax[w]);

  // Pass 2 over registers: sum exp, dot with true, sum of true.
  float se = 0.0f, dt = 0.0f, ss = 0.0f;
#pragma unroll
  for (int k = 0; k < MAXIT; ++k) {
    const v4f x = lx[k], tv = lt[k];
    se += __expf(x.x - lmax) + __expf(x.y - lmax) +
          __expf(x.z - lmax) + __expf(x.w - lmax);
    dt = fmaf(x.x, tv.x, fmaf(x.y, tv.y, fmaf(x.z, tv.z, fmaf(x.w, tv.w, dt))));
    ss += tv.x + tv.y + tv.z + tv.w;
  }

  // WMMA-based wave sums (uniform control flow here -> EXEC all ones).
  se = wave_sum_wmma(se);
  dt = wave_sum_wmma(dt);
  ss = wave_sum_wmma(ss);

  __shared__ float sred[NWAVE][3];
  if (lane == 0) { sred[wave][0] = se; sred[wave][1] = dt; sred[wave][2] = ss; }
  __syncthreads();
  if (tid == 0) {
    float SE = 0.0f, DT = 0.0f, S = 0.0f;
#pragma unroll
    for (int w = 0; w < NWAVE; ++w) {
      SE += sred[w][0]; DT += sred[w][1]; S += sred[w][2];
    }
    const float lse = lmax + __logf(SE);
    ws[row] = (lse * S - DT) * 0.1f;   // row contribution to ce (pre 1/B)
  }
}

// Single-block final combine: distorted[t], undistorted, elu terms, depressor.
__global__ __launch_bounds__(256) void final_kernel(
    const float* __restrict__ logit_var,
    const float* __restrict__ ws,
    float* __restrict__ out) {
  const int tid = threadIdx.x;
  __shared__ float sbuf[NWAVE];
  __shared__ float sdist[T_DIM];

  // Undistorted loss: mean over B of rows [T*B, T*B+B).
  float up = ws[T_DIM * B_DIM + tid] + ws[T_DIM * B_DIM + tid + 256];
  const float u = block_sum_256(up, sbuf) * (1.0f / (float)B_DIM);

  // Distorted mean per sample t.
  for (int t = 0; t < T_DIM; ++t) {
    float v = ws[t * B_DIM + tid] + ws[t * B_DIM + tid + 256];
    v = block_sum_256(v, sbuf);
    if (tid == 0) sdist[t] = v * (1.0f / (float)B_DIM);
  }

  // Variance depressor: mean(exp(logit_var) - 1) over B.
  float vd = (__expf(logit_var[tid]) - 1.0f) +
             (__expf(logit_var[tid + 256]) - 1.0f);
  vd = block_sum_256(vd, sbuf);

  if (tid == 0) {
    float gce = 0.0f, vl = 0.0f;
#pragma unroll
    for (int t = 0; t < T_DIM; ++t) {
      const float d = sdist[t];
      gce += d;
      const float x = u - d;                      // elu(x), alpha = 1
      const float e = (x > 0.0f) ? x : (__expf(x) - 1.0f);
      vl += -e;
    }
    out[0] = gce * (1.0f / (float)T_DIM);         // gce_loss
    out[1] = vl  * (1.0f / (float)T_DIM);         // variance_loss
    out[2] = u;                                   // undistorted_loss
    out[3] = vd  * (1.0f / (float)B_DIM);         // variance_depressor
  }
}

extern "C" void kernel_launch(void* const* d_in, const int* in_sizes, int n_in,
                              void* d_out, int out_size, void* d_ws, size_t ws_size,
                              hipStream_t stream) {
  const float* logit_var = (const float*)d_in[0];
  const float* pred      = (const float*)d_in[1];
  const float* tru       = (const float*)d_in[2];
  const float* noise     = (const float*)d_in[3];
  float* out = (float*)d_out;
  float* ws  = (float*)d_ws;   // needs (T+1)*B = 13312 floats

  const int nrows = (T_DIM + 1) * B_DIM;
  row_kernel<<<dim3(nrows), dim3(BLOCK), 0, stream>>>(logit_var, pred, tru,
                                                      noise, ws);
  final_kernel<<<dim3(1), dim3(256), 0, stream>>>(logit_var, ws, out);
}